// Track_Block_4664334483784
// MI455X (gfx1250) — compile-verified
//
#include <hip/hip_runtime.h>
#include <math.h>

typedef __attribute__((ext_vector_type(2))) float v2f;
typedef __attribute__((ext_vector_type(8))) float v8f;

#define HW_OUT 50176   // 224*224
#define HIDDEN 64
#define IN_F   7

// Fast silu: v * rcp(1 + exp(-v)).  v_exp_f32 + v_rcp_f32 are TRANS ops and
// co-execute with WMMA (both tracked as TRANS on CDNA5); avoids the
// correctly-rounded v_div_scale/v_div_fmas/v_div_fixup chain.
__device__ __forceinline__ float silu_f(float v) {
    return v * __builtin_amdgcn_rcpf(1.0f + __expf(-v));
}

__global__ void zero_acc_kernel(float* __restrict__ acc) {
    if (threadIdx.x < HIDDEN) acc[threadIdx.x] = 0.0f;
}

// Phase 1: gacc[j] = sum_n silu( (silu(x) @ W1 + b1)[n, j] )
// One wave processes 16-row tiles of x via V_WMMA_F32_16X16X4_F32.
// K=7 is split into K[0..3] (lo) and K[4..6]+zero pad (hi).
__global__ void __launch_bounds__(256)
phase1_kernel(const float* __restrict__ x,
              const float* __restrict__ W1,
              const float* __restrict__ b1,
              float* __restrict__ gacc,
              int N) {
    __shared__ float sacc[HIDDEN];
    const int tid = threadIdx.x;
    if (tid < HIDDEN) sacc[tid] = 0.0f;
    __syncthreads();

    const int lane = tid & 31;
    const int n    = lane & 15;   // B/C column within 16-wide tile; A row within tile
    const int g    = lane >> 4;   // half-wave group

    // B fragments of W1 (7x64, row-major). ISA layout: b[r] = B[r + 2g][n].
    // lo covers K=0..3 ; hi covers K=4..7 with K=7 zero-padded.
    v2f blo[4], bhi[4];
    float bias[4];
#pragma unroll
    for (int t = 0; t < 4; ++t) {
        const int col = t * 16 + n;
        blo[t].x = W1[(0 + 2 * g) * HIDDEN + col];   // K = 0 / 2
        blo[t].y = W1[(1 + 2 * g) * HIDDEN + col];   // K = 1 / 3
        bhi[t].x = W1[(4 + 2 * g) * HIDDEN + col];   // K = 4 / 6
        bhi[t].y = (g == 0) ? W1[5 * HIDDEN + col]   // K = 5
                            : 0.0f;                  // K = 7 (pad)
        bias[t]  = b1[col];
    }

    v8f acc[4];
#pragma unroll
    for (int t = 0; t < 4; ++t) acc[t] = {};

    const int waveGlobal = (blockIdx.x * blockDim.x + tid) >> 5;
    const int numWaves   = (gridDim.x * blockDim.x) >> 5;
    const int fullTiles  = N >> 4;   // main loop only touches complete 16-row tiles

    for (int tile = waveGlobal; tile < fullTiles; tile += numWaves) {
        // A fragment: a[r] = silu(x[row][2g + r]) (lo), silu(x[row][4 + 2g + r]) (hi)
        const float* xr = x + (long)(tile * 16 + n) * IN_F;
        v2f alo, ahi;
        alo.x = silu_f(xr[2 * g + 0]);
        alo.y = silu_f(xr[2 * g + 1]);
        ahi.x = silu_f(xr[4 + 2 * g]);
        ahi.y = (g == 0) ? silu_f(xr[5]) : 0.0f;  // K=7 pad for g==1

#pragma unroll
        for (int t = 0; t < 4; ++t) {
            v8f d = {};
            d = __builtin_amdgcn_wmma_f32_16x16x4_f32(
                    false, alo, false, blo[t], (short)0, d, false, false);
            d = __builtin_amdgcn_wmma_f32_16x16x4_f32(
                    false, ahi, false, bhi[t], (short)0, d, false, false);
#pragma unroll
            for (int r = 0; r < 8; ++r)
                acc[t][r] += silu_f(d[r] + bias[t]);
        }
    }

    // Reduce: lane's partial for column t*16+n = sum over its 8 D rows,
    // then fold the two half-waves, LDS-accumulate, then one global atomic.
#pragma unroll
    for (int t = 0; t < 4; ++t) {
        float p = 0.0f;
#pragma unroll
        for (int r = 0; r < 8; ++r) p += acc[t][r];
        p += __shfl_xor(p, 16, 32);
        if (g == 0) atomicAdd(&sacc[t * 16 + n], p);
    }
    __syncthreads();
    if (tid < HIDDEN) atomicAdd(&gacc[tid], sacc[tid]);

    // Tail rows (N not a multiple of 16): scalar path on one wave only.
    // Dead at runtime for N = 500000 but kept for correctness in general.
    const int tailStart = (N >> 4) << 4;
    if (blockIdx.x == 0 && tid < 32 && tailStart < N) {
        for (int c = lane; c < HIDDEN; c += 32) {
            float s = 0.0f;
            for (int row = tailStart; row < N; ++row) {
                const float* xr = x + (long)row * IN_F;
                float h = b1[c];
                for (int k = 0; k < IN_F; ++k)
                    h = fmaf(silu_f(xr[k]), W1[k * HIDDEN + c], h);
                s += silu_f(h);
            }
            atomicAdd(&gacc[c], s);
        }
    }
}

// Phase 2: out[i] = clamp(silu(mean_vec @ W2[:, i] + b2[i]), 0, 1)
// Bandwidth-bound on W2 (12.8 MB): float4 coalesced loads, m in LDS.
__global__ void __launch_bounds__(256)
phase2_kernel(const float* __restrict__ gacc,
              const float* __restrict__ W2,
              const float* __restrict__ b2,
              float* __restrict__ out,
              float invN) {
    __shared__ float m[HIDDEN];
    if (threadIdx.x < HIDDEN) m[threadIdx.x] = gacc[threadIdx.x] * invN;
    __syncthreads();

    const int i4 = (blockIdx.x * blockDim.x + threadIdx.x) * 4;
    if (i4 >= HW_OUT) return;

    const float4 bv = *(const float4*)(b2 + i4);
    float a0 = bv.x, a1 = bv.y, a2 = bv.z, a3 = bv.w;
#pragma unroll 8
    for (int j = 0; j < HIDDEN; ++j) {
        const float4 w = *(const float4*)(W2 + (size_t)j * HW_OUT + i4);
        const float mj = m[j];
        a0 = fmaf(mj, w.x, a0);
        a1 = fmaf(mj, w.y, a1);
        a2 = fmaf(mj, w.z, a2);
        a3 = fmaf(mj, w.w, a3);
    }
    float4 r;
    r.x = fminf(fmaxf(silu_f(a0), 0.0f), 1.0f);
    r.y = fminf(fmaxf(silu_f(a1), 0.0f), 1.0f);
    r.z = fminf(fmaxf(silu_f(a2), 0.0f), 1.0f);
    r.w = fminf(fmaxf(silu_f(a3), 0.0f), 1.0f);
    *(float4*)(out + i4) = r;
}

extern "C" void kernel_launch(void* const* d_in, const int* in_sizes, int n_in,
                              void* d_out, int out_size, void* d_ws, size_t ws_size,
                              hipStream_t stream) {
    const float* x  = (const float*)d_in[0];
    const float* W1 = (const float*)d_in[1];
    const float* b1 = (const float*)d_in[2];
    const float* W2 = (const float*)d_in[3];
    const float* b2 = (const float*)d_in[4];
    float* outp = (float*)d_out;
    float* gacc = (float*)d_ws;

    const int N = in_sizes[0] / IN_F;  // 500000

    zero_acc_kernel<<<1, 64, 0, stream>>>(gacc);
    phase1_kernel<<<256, 256, 0, stream>>>(x, W1, b1, gacc, N);

    const int vec_threads = HW_OUT / 4;  // 12544
    phase2_kernel<<<(vec_threads + 255) / 256, 256, 0, stream>>>(
        gacc, W2, b2, outp, 1.0f / (float)N);
}